// WLContinuous_7241314861278
// MI455X (gfx1250) — compile-verified
//
#include <hip/hip_runtime.h>

#define N_NODES 100000
#define N_EDGES 1600000
#define D_FEAT  64
#define D_OUT   128

typedef __attribute__((ext_vector_type(2))) float v2f;
typedef __attribute__((ext_vector_type(8))) float v8f;

// ---------------- utility ----------------
__global__ void zero_f32_kernel(float* __restrict__ p, int n) {
    int i = blockIdx.x * blockDim.x + threadIdx.x;
    if (i < n) p[i] = 0.0f;
}

// ---------------- degree ----------------
__global__ void degree_kernel(const int* __restrict__ dst, float* __restrict__ deg, int nE) {
    int e = blockIdx.x * blockDim.x + threadIdx.x;
    if (e < nE) atomicAdd(&deg[dst[e]], 1.0f);
}

__global__ void deg_inv_kernel(float* __restrict__ deg, int n) {
    int i = blockIdx.x * blockDim.x + threadIdx.x;
    if (i < n) deg[i] = 1.0f / fmaxf(deg[i], 1.0f);
}

// ---------------- edge scatter: agg[dst] += h[src] ----------------
// 16 threads per edge, each moves a float4 chunk (coalesced 16B gather,
// 4 global_atomic_add_f32 into L2-resident agg).
__global__ void scatter_kernel(const float* __restrict__ h,
                               const int* __restrict__ src,
                               const int* __restrict__ dst,
                               float* __restrict__ agg, int nE) {
    long long gid = (long long)blockIdx.x * blockDim.x + threadIdx.x;
    int e = (int)(gid >> 4);
    int c = ((int)gid & 15) * 4;
    if (e >= nE) return;
    int s = src[e];
    int d = dst[e];
    const float4 v = *(const float4*)(h + (long long)s * D_FEAT + c);
    float* ap = agg + (long long)d * D_FEAT + c;
    atomicAdd(ap + 0, v.x);
    atomicAdd(ap + 1, v.y);
    atomicAdd(ap + 2, v.z);
    atomicAdd(ap + 3, v.w);
}

// ---------------- layer0: h = relu(0.5*(x + agg*deg_inv)) ----------------
__global__ void combine0_kernel(const float* __restrict__ x,
                                const float* __restrict__ agg,
                                const float* __restrict__ dinv,
                                float* __restrict__ h, int n) {
    int i = blockIdx.x * blockDim.x + threadIdx.x;
    if (i >= n) return;
    int node = i >> 6;  // D_FEAT = 64
    float v = 0.5f * (x[i] + agg[i] * dinv[node]);
    h[i] = fmaxf(v, 0.0f);
}

// ---------------- residual layers: h = h + relu(0.5*(h + agg*deg_inv)) ----------------
__global__ void combine_res_kernel(float* __restrict__ h,
                                   const float* __restrict__ agg,
                                   const float* __restrict__ dinv, int n) {
    int i = blockIdx.x * blockDim.x + threadIdx.x;
    if (i >= n) return;
    int node = i >> 6;
    float hv = h[i];
    float v = 0.5f * (hv + agg[i] * dinv[node]);
    h[i] = hv + fmaxf(v, 0.0f);
}

// ---------------- final GEMM: out = h @ W + b, fp32 WMMA 16x16x4 ----------------
// One wave per 16x16 output tile. K=64 -> 16 v_wmma_f32_16x16x4_f32 per tile.
// A layout (16x4 f32): lanes 0-15 -> M=lane, K={k0,k0+1}; lanes 16-31 -> K={k0+2,k0+3}.
// B layout (4x16 f32): lanes 0-15 -> N=lane, K={k0,k0+1}; lanes 16-31 -> K={k0+2,k0+3}.
// D layout (16x16 f32): VGPR r -> M=r (lanes 0-15) / M=8+r (lanes 16-31), N=lane&15.
__global__ void gemm_wmma_kernel(const float* __restrict__ h,
                                 const float* __restrict__ W,
                                 const float* __restrict__ bias,
                                 float* __restrict__ out) {
    int wave = (blockIdx.x * blockDim.x + threadIdx.x) >> 5;  // global wave id
    int lane = threadIdx.x & 31;

    // tiles: (N_NODES/16)=6250 M-tiles x (D_OUT/16)=8 N-tiles
    int mt = wave >> 3;
    int nt = wave & 7;
    int m0 = mt * 16;
    int n0 = nt * 16;

    int l15   = lane & 15;
    int khalf = (lane >> 4) * 2;  // 0 or 2

    const float* hrow = h + (long long)(m0 + l15) * D_FEAT;
    const float* wcol = W + (n0 + l15);

    v8f c = {};
#pragma unroll
    for (int k0 = 0; k0 < D_FEAT; k0 += 4) {
        v2f a, bm;
        a.x  = hrow[k0 + khalf];
        a.y  = hrow[k0 + khalf + 1];
        bm.x = wcol[(long long)(k0 + khalf) * D_OUT];
        bm.y = wcol[(long long)(k0 + khalf + 1) * D_OUT];
        c = __builtin_amdgcn_wmma_f32_16x16x4_f32(false, a, false, bm,
                                                  (short)0, c, false, false);
    }

    float bv = bias[n0 + l15];
    int mbase = m0 + (lane >> 4) * 8;
#pragma unroll
    for (int r = 0; r < 8; ++r) {
        out[(long long)(mbase + r) * D_OUT + n0 + l15] = c[r] + bv;
    }
}

// ---------------- launcher ----------------
extern "C" void kernel_launch(void* const* d_in, const int* in_sizes, int n_in,
                              void* d_out, int out_size, void* d_ws, size_t ws_size,
                              hipStream_t stream) {
    const float* x  = (const float*)d_in[0];            // [N_NODES, 64]
    const int*   ei = (const int*)d_in[1];              // [2, N_EDGES]
    const float* W  = (const float*)d_in[2];            // [64, 128]
    const float* b  = (const float*)d_in[3];            // [128]
    float* out = (float*)d_out;                         // [N_NODES, 128]

    const int* src = ei;
    const int* dst = ei + N_EDGES;

    // workspace layout
    const long long NF = (long long)N_NODES * D_FEAT;   // 6.4M floats
    float* h    = (float*)d_ws;                         // [N_NODES*64]
    float* agg  = h + NF;                               // [N_NODES*64]
    float* dinv = agg + NF;                             // [N_NODES]

    const int B = 256;
    const int gridNF    = (int)((NF + B - 1) / B);              // 25000
    const int gridNodes = (N_NODES + B - 1) / B;                // 391
    const int gridEdges = (N_EDGES + B - 1) / B;                // 6250
    const int gridScat  = (int)(((long long)N_EDGES * 16 + B - 1) / B);  // 100000

    // degree -> deg_inv (once)
    zero_f32_kernel<<<gridNodes, B, 0, stream>>>(dinv, N_NODES);
    degree_kernel<<<gridEdges, B, 0, stream>>>(dst, dinv, N_EDGES);
    deg_inv_kernel<<<gridNodes, B, 0, stream>>>(dinv, N_NODES);

    // layer 0: agg(x) -> h = relu(0.5*(x + agg*dinv))
    zero_f32_kernel<<<gridNF, B, 0, stream>>>(agg, (int)NF);
    scatter_kernel<<<gridScat, B, 0, stream>>>(x, src, dst, agg, N_EDGES);
    combine0_kernel<<<gridNF, B, 0, stream>>>(x, agg, dinv, h, (int)NF);

    // layers 1..3: residual WL
    for (int l = 1; l < 4; ++l) {
        zero_f32_kernel<<<gridNF, B, 0, stream>>>(agg, (int)NF);
        scatter_kernel<<<gridScat, B, 0, stream>>>(h, src, dst, agg, N_EDGES);
        combine_res_kernel<<<gridNF, B, 0, stream>>>(h, agg, dinv, (int)NF);
    }

    // final linear: out = h @ W + b  (fp32 WMMA)
    // 50000 tiles, 8 waves/block -> 6250 blocks
    gemm_wmma_kernel<<<(N_NODES / 16) * (D_OUT / 16) / 8, B, 0, stream>>>(h, W, b, out);
}